// MaskedNorm_6940667150729
// MI455X (gfx1250) — compile-verified
//
#include <hip/hip_runtime.h>

// MI455X / gfx1250: wave32, WMMA 16x16x4 f32 path for the masked column reduction.

typedef float v2f __attribute__((ext_vector_type(2)));
typedef float v4f __attribute__((ext_vector_type(4)));
typedef float v8f __attribute__((ext_vector_type(8)));

#define NB      16
#define NT      4096
#define NF      512          // features
#define NROWS   (NB * NT)    // 65536 rows
#define NTILES  (NF / 16)    // 32 feature tiles of 16
#define NRB     32           // row blocks
#define RPB     (NROWS / NRB)   // 2048 rows per block
#define MN_EPS  1e-5f

// ---------------------------------------------------------------------------
// Pass 1: per (feature-tile, row-block) partial sums of w*x and w*x^2 via WMMA.
//   D = A(16x4) * B(4x16) + C  with A[m][k] = w[row k] (rows identical)
//   => every row of D = masked column sums of the 4-row tile of x.
// A layout (f32 16x4):  lanes 0-15: VGPR0=K0, VGPR1=K1 ; lanes 16-31: K2,K3
// B layout (f32 4x16):  VGPR0: K=(half?2:0), VGPR1: K+1 ; N = lane&15
// ---------------------------------------------------------------------------
__global__ void __launch_bounds__(256)
mn_reduce_kernel(const float* __restrict__ x,
                 const int*   __restrict__ mask,
                 float* __restrict__ psum,    // [NRB][NF]
                 float* __restrict__ psq,     // [NRB][NF]
                 float* __restrict__ pcount)  // [NRB]
{
    const int tile = blockIdx.x & (NTILES - 1);
    const int rb   = blockIdx.x >> 5;            // NTILES == 32
    const int tid  = threadIdx.x;
    const int wave = tid >> 5;
    const int lane = tid & 31;
    const int half = lane >> 4;                  // 0 -> K0/K1, 1 -> K2/K3
    const int n    = lane & 15;
    const int f    = tile * 16 + n;
    const int rowBase = rb * RPB;

    v8f csum = {};
    v8f csq  = {};

    // 8 waves x 4 rows = 32 rows per iteration; RPB/32 = 64 iterations.
    #pragma unroll 4
    for (int it = 0; it < RPB / 32; ++it) {
        const int r0 = rowBase + (it * 8 + wave) * 4 + half * 2;
        const int r1 = r0 + 1;

        const float w0 = (mask[r0] > 0) ? 1.0f : 0.0f;
        const float w1 = (mask[r1] > 0) ? 1.0f : 0.0f;
        const float b0 = x[(size_t)r0 * NF + f];
        const float b1 = x[(size_t)r1 * NF + f];

        v2f Am = {w0, w1};           // mask weights (all 16 rows of A identical)
        v2f Bx = {b0, b1};           // x tile
        v2f Bq = {b0 * b0, b1 * b1}; // x^2 tile (w applied by A)

        csum = __builtin_amdgcn_wmma_f32_16x16x4_f32(
            false, Am, false, Bx, (short)0, csum, false, false);
        csq  = __builtin_amdgcn_wmma_f32_16x16x4_f32(
            false, Am, false, Bq, (short)0, csq, false, false);
    }

    // Every row of D is identical => c[0] on lane L is colsum[L&15].
    __shared__ float s_sum[8][16];
    __shared__ float s_sq [8][16];
    if (lane < 16) {
        s_sum[wave][lane] = csum[0];
        s_sq [wave][lane] = csq[0];
    }
    __syncthreads();

    if (tid < 16) {
        float a = 0.0f, b = 0.0f;
        #pragma unroll
        for (int w = 0; w < 8; ++w) {  // fixed order -> deterministic
            a += s_sum[w][tid];
            b += s_sq [w][tid];
        }
        psum[(size_t)rb * NF + tile * 16 + tid] = a;
        psq [(size_t)rb * NF + tile * 16 + tid] = b;
    }

    // One tile per row-block also produces the valid-row count (fixed-order tree).
    if (tile == 0) {
        int c = 0;
        #pragma unroll
        for (int j = 0; j < RPB / 256; ++j)      // 8 rows per thread
            c += (mask[rowBase + j * 256 + tid] > 0) ? 1 : 0;

        __shared__ int s_cnt[256];
        s_cnt[tid] = c;
        __syncthreads();
        for (int s = 128; s > 0; s >>= 1) {
            if (tid < s) s_cnt[tid] += s_cnt[tid + s];
            __syncthreads();
        }
        if (tid == 0) pcount[rb] = (float)s_cnt[0];
    }
}

// ---------------------------------------------------------------------------
// Pass 2: fold the 32 partials per feature, emit scale/shift.
//   normed = x*scale + shift,  scale = rsqrt(var+eps)*gamma,
//   shift = beta - mean*scale
// ---------------------------------------------------------------------------
__global__ void __launch_bounds__(NF)
mn_finalize_kernel(const float* __restrict__ psum,
                   const float* __restrict__ psq,
                   const float* __restrict__ pcount,
                   const float* __restrict__ gamma,
                   const float* __restrict__ beta,
                   float* __restrict__ scl,
                   float* __restrict__ sft)
{
    const int f = threadIdx.x;  // 512 threads
    float n = 0.0f;
    #pragma unroll
    for (int i = 0; i < NRB; ++i) n += pcount[i];
    n = fmaxf(n, 1.0f);

    float s = 0.0f, q = 0.0f;
    #pragma unroll
    for (int i = 0; i < NRB; ++i) {
        s += psum[(size_t)i * NF + f];
        q += psq [(size_t)i * NF + f];
    }
    const float mean = s / n;
    const float var  = fmaxf(q / n - mean * mean, 0.0f);  // biased variance
    const float inv  = rsqrtf(var + MN_EPS);
    const float a    = inv * gamma[f];
    scl[f] = a;
    sft[f] = beta[f] - mean * a;
}

// ---------------------------------------------------------------------------
// Pass 3: elementwise apply. b128 loads; x should still be L2-resident (134 MB
// < 192 MB L2) from pass 1; out is written non-temporal (never re-read).
// ---------------------------------------------------------------------------
__global__ void __launch_bounds__(256)
mn_apply_kernel(const float* __restrict__ x,
                const int*   __restrict__ mask,
                const float* __restrict__ scl,
                const float* __restrict__ sft,
                float* __restrict__ out)
{
    const size_t i4 = (size_t)blockIdx.x * blockDim.x + threadIdx.x;
    const size_t e  = i4 * 4;                 // 4 consecutive features, same row
    const int row   = (int)(e >> 9);          // NF = 512
    const int f     = (int)(e & (NF - 1));    // multiple of 4

    const v4f xv = *(const v4f*)(x   + e);
    const v4f av = *(const v4f*)(scl + f);
    const v4f bv = *(const v4f*)(sft + f);
    const bool m = mask[row] > 0;

    v4f o;
    o.x = m ? fmaf(xv.x, av.x, bv.x) : xv.x;
    o.y = m ? fmaf(xv.y, av.y, bv.y) : xv.y;
    o.z = m ? fmaf(xv.z, av.z, bv.z) : xv.z;
    o.w = m ? fmaf(xv.w, av.w, bv.w) : xv.w;

    __builtin_nontemporal_store(o, (v4f*)(out + e));
}

// ---------------------------------------------------------------------------
extern "C" void kernel_launch(void* const* d_in, const int* in_sizes, int n_in,
                              void* d_out, int out_size, void* d_ws, size_t ws_size,
                              hipStream_t stream)
{
    (void)in_sizes; (void)n_in; (void)out_size; (void)ws_size;

    const float* x     = (const float*)d_in[0];
    const int*   mask  = (const int*)  d_in[1];
    const float* gamma = (const float*)d_in[2];
    const float* beta  = (const float*)d_in[3];
    float*       out   = (float*)d_out;

    // ws layout (floats): psum[NRB*NF] | psq[NRB*NF] | pcount[NRB] | scl[NF] | sft[NF]
    float* ws     = (float*)d_ws;
    float* psum   = ws;
    float* psq    = psum + (size_t)NRB * NF;
    float* pcount = psq  + (size_t)NRB * NF;
    float* scl    = pcount + NRB;
    float* sft    = scl + NF;

    mn_reduce_kernel<<<NTILES * NRB, 256, 0, stream>>>(x, mask, psum, psq, pcount);
    mn_finalize_kernel<<<1, NF, 0, stream>>>(psum, psq, pcount, gamma, beta, scl, sft);

    const size_t nvec = (size_t)NROWS * NF / 4;       // 8,388,608 float4s
    mn_apply_kernel<<<(int)(nvec / 256), 256, 0, stream>>>(x, mask, scl, sft, out);
}